// GRUModule_9148280340872
// MI455X (gfx1250) — compile-verified
//
#include <hip/hip_runtime.h>
#include <math.h>

// ---------------------------------------------------------------------------
// Shapes (fixed by the reference)
// ---------------------------------------------------------------------------
#define Wt   64            // timesteps
#define Bb   32
#define Ii   128
#define Nn   (Bb * Ii)     // 4096 sequences
#define Ff   128           // feature dim (K of the GEMM)
#define Hh   136           // 8 + F
#define ROWS ((size_t)Wt * Nn)   // 262144 GEMM rows
#define TILES (ROWS / 16)        // 16384 row-tiles of 16

typedef __attribute__((ext_vector_type(2))) float v2f;
typedef __attribute__((ext_vector_type(8))) float v8f;

// Workspace layout (floats):
//   [0      .. 2047]  W_eff padded to 16 x 128 (rows 3..15 = 0)
//   [2048   .. 2063]  b_eff padded to 16
//   [2112   .. ]      gi : (W*N) x 3  = 786432 floats
#define WS_WEFF 0
#define WS_BEFF 2048
#define WS_GI   2112

// ---------------------------------------------------------------------------
// Kernel 1: fuse the two linear layers.
//   W_eff[g][f] = sum_h W_ih[g][h] * W_lin[h][f]      (g<3; g in 3..15 -> 0)
//   b_eff[g]    = sum_h W_ih[g][h] * b_lin[h] + b_ih[g]
// ---------------------------------------------------------------------------
__global__ void prep_kernel(const float* __restrict__ W_lin,
                            const float* __restrict__ b_lin,
                            const float* __restrict__ W_ih,
                            const float* __restrict__ b_ih,
                            float* __restrict__ ws) {
    int f = threadIdx.x;              // 0..127
    float acc0 = 0.f, acc1 = 0.f, acc2 = 0.f;
    for (int h = 0; h < Hh; ++h) {
        float wl = W_lin[h * Ff + f];
        acc0 = fmaf(W_ih[0 * Hh + h], wl, acc0);
        acc1 = fmaf(W_ih[1 * Hh + h], wl, acc1);
        acc2 = fmaf(W_ih[2 * Hh + h], wl, acc2);
    }
    float* Weff = ws + WS_WEFF;
    Weff[0 * Ff + f] = acc0;
    Weff[1 * Ff + f] = acc1;
    Weff[2 * Ff + f] = acc2;
    for (int g = 3; g < 16; ++g) Weff[g * Ff + f] = 0.f;   // zero-pad rows

    if (f < 16) {
        float b = 0.f;
        if (f < 3) {
            float s = 0.f;
            for (int h = 0; h < Hh; ++h) s = fmaf(W_ih[f * Hh + h], b_lin[h], s);
            b = s + b_ih[f];
        }
        ws[WS_BEFF + f] = b;
    }
}

// ---------------------------------------------------------------------------
// Kernel 2: gi[row][gate] = sum_k x[row][k] * W_eff[gate][k] + b_eff[gate]
// via V_WMMA_F32_16X16X4_F32.  One wave per 16-row tile, 32 wmma ops (K=128).
//
// f32 16x4 A layout: lanes 0-15 hold M=0..15 / K={k,k+1}; lanes 16-31 hold
// M=0..15 / K={k+2,k+3}.  B (4x16) mirrors with N = lane%16.
// C/D: VGPR r, lane L -> M = r + 8*(L>>4), N = L%16.
// ---------------------------------------------------------------------------
__global__ void gi_kernel(const float* __restrict__ x,
                          const float* __restrict__ ws,
                          float* __restrict__ gi) {
    const int lane  = threadIdx.x & 31;
    const int wave  = threadIdx.x >> 5;
    const size_t tile = (size_t)blockIdx.x * 8 + wave;   // 8 waves / block

    const int m     = lane & 15;      // A row within tile / B column (gate)
    const int khalf = lane >> 4;      // 0 or 1 -> K offset 0 or 2

    const float* __restrict__ rowp = x + (tile * 16 + m) * Ff;  // this lane's A row
    const float* __restrict__ Weff = ws + WS_WEFF;              // 16x128, zero-padded
    const float* __restrict__ beff = ws + WS_BEFF;

    v8f c = {};
    #pragma unroll 4
    for (int kb = 0; kb < Ff; kb += 4) {
        const int k0 = kb + 2 * khalf;
        v2f a, b;
        a.x = rowp[k0];
        a.y = rowp[k0 + 1];
        b.x = Weff[m * Ff + k0];        // B[k][n] = W_eff[n][k], n = lane%16
        b.y = Weff[m * Ff + k0 + 1];
        c = __builtin_amdgcn_wmma_f32_16x16x4_f32(
                /*neg_a=*/false, a, /*neg_b=*/false, b,
                /*c_mod=*/(short)0, c, /*reuse_a=*/false, /*reuse_b=*/false);
    }

    // Scatter the 3 live gate columns (+bias) to gi[(row)*3 + gate].
    if (m < 3) {
        const float bias = beff[m];
        const size_t rbase = tile * 16 + 8 * khalf;
        #pragma unroll
        for (int r = 0; r < 8; ++r) {
            gi[(rbase + r) * 3 + m] = c[r] + bias;
        }
    }
}

// ---------------------------------------------------------------------------
// Kernel 3: sequential GRU scan.  One thread per sequence (N=4096).
//   r = sigmoid(gi_r + W_hh0*h + b_hh0)
//   z = sigmoid(gi_z + W_hh1*h + b_hh1)
//   n = tanh  (gi_n + r*(W_hh2*h + b_hh2))
//   h = (1-z)*n + z*h ;  out[w][seq] = h
// ---------------------------------------------------------------------------
__global__ void scan_kernel(const float* __restrict__ gi,
                            const float* __restrict__ state,
                            const float* __restrict__ W_hh,
                            const float* __restrict__ b_hh,
                            float* __restrict__ out) {
    const int n = blockIdx.x * blockDim.x + threadIdx.x;   // 0..4095
    float h = state[n];
    const float wr = W_hh[0], wz = W_hh[1], wn = W_hh[2];
    const float br = b_hh[0], bz = b_hh[1], bn = b_hh[2];

    for (int w = 0; w < Wt; ++w) {
        const size_t base = ((size_t)w * Nn + n) * 3;
        const float gr = gi[base + 0];
        const float gz = gi[base + 1];
        const float gn = gi[base + 2];
        const float r  = 1.f / (1.f + expf(-(gr + wr * h + br)));
        const float z  = 1.f / (1.f + expf(-(gz + wz * h + bz)));
        const float nv = tanhf(gn + r * (wn * h + bn));
        h = (1.f - z) * nv + z * h;
        out[(size_t)w * Nn + n] = h;
    }
}

// ---------------------------------------------------------------------------
extern "C" void kernel_launch(void* const* d_in, const int* in_sizes, int n_in,
                              void* d_out, int out_size, void* d_ws, size_t ws_size,
                              hipStream_t stream) {
    const float* inputs = (const float*)d_in[0];  // (64,32,128,128)
    const float* state  = (const float*)d_in[1];  // (1,32,128,1)
    const float* W_lin  = (const float*)d_in[2];  // (136,128)
    const float* b_lin  = (const float*)d_in[3];  // (136)
    const float* W_ih   = (const float*)d_in[4];  // (3,136)
    const float* b_ih   = (const float*)d_in[5];  // (3)
    const float* W_hh   = (const float*)d_in[6];  // (3)
    const float* b_hh   = (const float*)d_in[7];  // (3)
    float* out = (float*)d_out;                   // (64,32,128,1)
    float* ws  = (float*)d_ws;
    float* gi  = ws + WS_GI;                      // (W*N,3) fp32 in workspace

    prep_kernel<<<1, 128, 0, stream>>>(W_lin, b_lin, W_ih, b_ih, ws);

    // 16384 tiles, 8 waves (tiles) per 256-thread block -> 2048 blocks.
    gi_kernel<<<(int)(TILES / 8), 256, 0, stream>>>(inputs, ws, gi);

    // 4096 sequences -> 16 blocks of 256.
    scan_kernel<<<Nn / 256, 256, 0, stream>>>(gi, state, W_hh, b_hh, out);
}